// projection_49795850830417
// MI455X (gfx1250) — compile-verified
//
#include <hip/hip_runtime.h>
#include <math.h>

// Problem constants (from reference)
#define N_PTS   8192
#define D_IN    10
#define CH      6
#define HID     16
#define MAX_SQ  12
#define TAYLOR_K 12
#define WAVES_PER_BLOCK 4

typedef float v2f __attribute__((ext_vector_type(2)));
typedef float v8f __attribute__((ext_vector_type(8)));

// ---------------------------------------------------------------------------
// Kernel 1: Aw = frame(A) = tril(A,-1) - tril(A,-1)^H   (skew-Hermitian)
// A, Aw stored as interleaved complex (float2), shape (D_IN, CH, 16, 16)
// ---------------------------------------------------------------------------
__global__ void frame_kernel(const float2* __restrict__ A, float2* __restrict__ Aw) {
    int t = blockIdx.x * blockDim.x + threadIdx.x;
    if (t >= D_IN * CH * HID * HID) return;
    int j = t & 15;
    int i = (t >> 4) & 15;
    int mat = t >> 8;
    const float2* Am = A + (size_t)mat * 256;
    float2 l  = (i > j) ? Am[i * 16 + j] : make_float2(0.f, 0.f); // L[i][j]
    float2 lt = (j > i) ? Am[j * 16 + i] : make_float2(0.f, 0.f); // L[j][i] = (L^T)[i][j]
    // Aw[i][j] = L[i][j] - conj(L[j][i])
    Aw[t] = make_float2(l.x - lt.x, l.y + lt.y);
}

// ---------------------------------------------------------------------------
// Complex 16x16 matmul D = A * B using V_WMMA_F32_16X16X4_F32.
// A-operands are held in registers (4 K-chunks of v2f, re / im / -im).
// B-operands are fetched from the wave's LDS tile (re/im planes, row-major).
//
// WMMA f32 operand layouts (wave32):
//   A (16x4): lane l<16 -> row l, K = {4q, 4q+1}; lane l>=16 -> row l-16, K = {4q+2, 4q+3}
//   B (4x16): lane l<16 -> col l, K = {4q, 4q+1}; lane l>=16 -> col l-16, K = {4q+2, 4q+3}
//   C/D (16x16): vgpr v, lane l -> row v + 8*(l/16), col l%16
// ---------------------------------------------------------------------------
__device__ __forceinline__ void cmatmul16(const v2f* __restrict__ Ar,
                                          const v2f* __restrict__ Ai,
                                          const v2f* __restrict__ Ain,
                                          const float* __restrict__ bre,
                                          const float* __restrict__ bim,
                                          int half, int col,
                                          v8f& Dre, v8f& Dim) {
    v2f Br[4], Bi[4];
#pragma unroll
    for (int q = 0; q < 4; ++q) {
        int k0 = 4 * q + 2 * half;
        Br[q][0] = bre[(k0    ) * 16 + col];
        Br[q][1] = bre[(k0 + 1) * 16 + col];
        Bi[q][0] = bim[(k0    ) * 16 + col];
        Bi[q][1] = bim[(k0 + 1) * 16 + col];
    }
    v8f accr = 0.0f;
    v8f acci = 0.0f;
    // Re(D) = Ar*Br - Ai*Bi  (chained through the accumulator, -Ai pre-negated)
#pragma unroll
    for (int q = 0; q < 4; ++q)
        accr = __builtin_amdgcn_wmma_f32_16x16x4_f32(false, Ar[q],  false, Br[q], (short)0, accr, false, false);
#pragma unroll
    for (int q = 0; q < 4; ++q)
        accr = __builtin_amdgcn_wmma_f32_16x16x4_f32(false, Ain[q], false, Bi[q], (short)0, accr, false, false);
    // Im(D) = Ar*Bi + Ai*Br
#pragma unroll
    for (int q = 0; q < 4; ++q)
        acci = __builtin_amdgcn_wmma_f32_16x16x4_f32(false, Ar[q],  false, Bi[q], (short)0, acci, false, false);
#pragma unroll
    for (int q = 0; q < 4; ++q)
        acci = __builtin_amdgcn_wmma_f32_16x16x4_f32(false, Ai[q],  false, Br[q], (short)0, acci, false, false);
    Dre = accr;
    Dim = acci;
}

// ---------------------------------------------------------------------------
// Kernel 2: one wave per (n,c) matrix.
//   AX = sum_d dX[n,d] * Aw[d,c]          (into WMMA C/D layout registers)
//   s  = ceil(log2(||AX||_1)) if > 1      (wave-uniform via shuffles)
//   T  = expm(AX * 2^-s) via 12-term Taylor (Horner), matmuls on WMMA
//   T  = T^2, min(s,12) times
// ---------------------------------------------------------------------------
__global__ __launch_bounds__(32 * WAVES_PER_BLOCK)
void expm_kernel(const float* __restrict__ dX,
                 const float2* __restrict__ Aw,
                 float2* __restrict__ out) {
    __shared__ float lds_re[WAVES_PER_BLOCK][256];
    __shared__ float lds_im[WAVES_PER_BLOCK][256];

    const int lane = threadIdx.x & 31;
    const int wave = threadIdx.x >> 5;
    const int mat  = blockIdx.x * WAVES_PER_BLOCK + wave;   // 0 .. N*CH-1
    const int n    = mat / CH;
    const int c    = mat % CH;

    const int half = lane >> 4;     // which half of the wave
    const int col  = lane & 15;     // C/D layout column (also A/B lane row/col)

    float* myre = lds_re[wave];
    float* myim = lds_im[wave];

    // ---- AX = sum_d dX[n,d] * Aw[d,c], directly in C/D layout ----
    float ax_re[8], ax_im[8];
#pragma unroll
    for (int v = 0; v < 8; ++v) { ax_re[v] = 0.f; ax_im[v] = 0.f; }
#pragma unroll
    for (int d = 0; d < D_IN; ++d) {
        float w = dX[n * D_IN + d];
        const float2* M = Aw + ((size_t)d * CH + c) * 256;
#pragma unroll
        for (int v = 0; v < 8; ++v) {
            int i = v + 8 * half;
            float2 e = M[i * 16 + col];
            ax_re[v] = fmaf(w, e.x, ax_re[v]);
            ax_im[v] = fmaf(w, e.y, ax_im[v]);
        }
    }

    // ---- induced 1-norm: max_j sum_i |AX[i,j]| ----
    float part = 0.f;
#pragma unroll
    for (int v = 0; v < 8; ++v)
        part += sqrtf(ax_re[v] * ax_re[v] + ax_im[v] * ax_im[v]);
    part += __shfl_xor(part, 16, 32);            // merge the two row-halves of column `col`
    float nm = part;
#pragma unroll
    for (int off = 8; off >= 1; off >>= 1)
        nm = fmaxf(nm, __shfl_xor(nm, off, 32)); // max over 16 columns (wave-uniform result)

    int s = 0;
    if (nm > 1.0f) s = (int)ceilf(log2f(nm));
    if (s < 0) s = 0;
    const float scale = exp2f(-(float)s);
    const int nsq = (s > MAX_SQ) ? MAX_SQ : s;   // reference squares only MAX_SQ times

    // ---- scale AX, stage to LDS, build A-operands (fixed for the Horner chain) ----
#pragma unroll
    for (int v = 0; v < 8; ++v) {
        ax_re[v] *= scale;
        ax_im[v] *= scale;
        int i = v + 8 * half;
        myre[i * 16 + col] = ax_re[v];
        myim[i * 16 + col] = ax_im[v];
    }
    asm volatile("s_wait_dscnt 0" ::: "memory");

    v2f A_re[4], A_im[4], A_imn[4];
#pragma unroll
    for (int q = 0; q < 4; ++q) {
        int k0 = 4 * q + 2 * half;
        int r  = col;                    // A-operand row for this lane
        A_re[q][0] = myre[r * 16 + k0];
        A_re[q][1] = myre[r * 16 + k0 + 1];
        A_im[q][0] = myim[r * 16 + k0];
        A_im[q][1] = myim[r * 16 + k0 + 1];
        A_imn[q]   = -A_im[q];
    }

    // ---- T = expm(AX*scale): Horner  T <- I + (AX_s * T)/k, k = K..1 ----
    // Fully unrolled: 1/k folds to an inline literal (no runtime fp division),
    // and the ds-store -> ds-load -> wmma chains software-pipeline across steps.
    v8f Tre, Tim;
#pragma unroll
    for (int v = 0; v < 8; ++v) {
        int i = v + 8 * half;
        Tre[v] = (i == col) ? 1.0f : 0.0f;
        Tim[v] = 0.0f;
    }
#pragma unroll
    for (int k = TAYLOR_K; k >= 1; --k) {
#pragma unroll
        for (int v = 0; v < 8; ++v) {
            int i = v + 8 * half;
            myre[i * 16 + col] = Tre[v];
            myim[i * 16 + col] = Tim[v];
        }
        asm volatile("s_wait_dscnt 0" ::: "memory");
        v8f Dre, Dim;
        cmatmul16(A_re, A_im, A_imn, myre, myim, half, col, Dre, Dim);
        const float inv = 1.0f / (float)k;   // compile-time constant after unroll
#pragma unroll
        for (int v = 0; v < 8; ++v) {
            int i = v + 8 * half;
            Tre[v] = fmaf(Dre[v], inv, (i == col) ? 1.0f : 0.0f);
            Tim[v] = Dim[v] * inv;
        }
    }

    // ---- repeated squaring: T <- T*T, nsq times (runtime, wave-uniform count) ----
    for (int it = 0; it < nsq; ++it) {
#pragma unroll
        for (int v = 0; v < 8; ++v) {
            int i = v + 8 * half;
            myre[i * 16 + col] = Tre[v];
            myim[i * 16 + col] = Tim[v];
        }
        asm volatile("s_wait_dscnt 0" ::: "memory");
        v2f Sr[4], Si[4], Sin[4];
#pragma unroll
        for (int q = 0; q < 4; ++q) {
            int k0 = 4 * q + 2 * half;
            int r  = col;
            Sr[q][0] = myre[r * 16 + k0];
            Sr[q][1] = myre[r * 16 + k0 + 1];
            Si[q][0] = myim[r * 16 + k0];
            Si[q][1] = myim[r * 16 + k0 + 1];
            Sin[q]   = -Si[q];
        }
        v8f Dre, Dim;
        cmatmul16(Sr, Si, Sin, myre, myim, half, col, Dre, Dim);
        Tre = Dre;
        Tim = Dim;
    }

    // ---- write result (complex64 interleaved) ----
    float2* o = out + (size_t)mat * 256;
#pragma unroll
    for (int v = 0; v < 8; ++v) {
        int i = v + 8 * half;
        o[i * 16 + col] = make_float2(Tre[v], Tim[v]);
    }
}

// ---------------------------------------------------------------------------
extern "C" void kernel_launch(void* const* d_in, const int* in_sizes, int n_in,
                              void* d_out, int out_size, void* d_ws, size_t ws_size,
                              hipStream_t stream) {
    const float*  dX = (const float*)d_in[0];           // (N, D_IN) float32
    const float2* A  = (const float2*)d_in[1];          // (D_IN, CH, 16, 16) complex64
    float2* Aw  = (float2*)d_ws;                        // (D_IN, CH, 16, 16) complex64 scratch
    float2* out = (float2*)d_out;                       // (N, CH, 16, 16) complex64

    const int frame_elems = D_IN * CH * HID * HID;      // 15360
    frame_kernel<<<(frame_elems + 255) / 256, 256, 0, stream>>>(A, Aw);

    const int nmat = N_PTS * CH;                        // 49152 waves
    expm_kernel<<<nmat / WAVES_PER_BLOCK, 32 * WAVES_PER_BLOCK, 0, stream>>>(dX, Aw, out);
}